// PAICM_83485574300357
// MI455X (gfx1250) — compile-verified
//
#include <hip/hip_runtime.h>
#include <hip/hip_bf16.h>

#define INPUT_DIM 362
#define HIDDEN    256
#define STYLE_NUM 60
#define OUTFIT_NUM 49710
#define BATCH     16384
#define DIM2      (2 * INPUT_DIM)       // 724
#define TILE_ELEMS (16 * INPUT_DIM)     // 5792 floats, contiguous 16-row tile

typedef __attribute__((ext_vector_type(2))) float v2f;
typedef __attribute__((ext_vector_type(8))) float v8f;
typedef unsigned int u32x4 __attribute__((ext_vector_type(4)));
typedef int i32x4 __attribute__((ext_vector_type(4)));
typedef int i32x8 __attribute__((ext_vector_type(8)));

#if defined(__has_builtin)
#if __has_builtin(__builtin_amdgcn_tensor_load_to_lds) && __has_builtin(__builtin_amdgcn_s_wait_tensorcnt)
#define USE_TDM_PATH 1
#endif
#endif

__device__ __forceinline__ v8f wmma4(v2f a, v2f b, v8f c) {
  // D = A(16x4 f32) * B(4x16 f32) + C(16x16 f32)
  return __builtin_amdgcn_wmma_f32_16x16x4_f32(
      false, a, false, b, (short)0, c, false, false);
}

__device__ __forceinline__ float sigm(float x) { return 1.f / (1.f + __expf(-x)); }
// -log_sigmoid(x) = softplus(-x)
__device__ __forceinline__ float softplus(float z) {
  return fmaxf(z, 0.f) + log1pf(__expf(-fabsf(z)));
}

// Issue one TDM copy of a contiguous TILE_ELEMS-float block: global -> LDS.
// Must be executed by exactly one wave; follow with s_wait_tensorcnt + barrier.
__device__ __forceinline__ void tdm_issue_tile(float* lds_dst, const float* src) {
#ifdef USE_TDM_PATH
  unsigned lds_off = (unsigned)(size_t)(void*)lds_dst;   // low 32 bits = LDS offset
  unsigned long long ga = (unsigned long long)(size_t)(const void*)src;
  u32x4 g0;
  g0[0] = 1u;                                            // count=1 (valid descriptor)
  g0[1] = lds_off;                                       // lds_addr
  g0[2] = (unsigned)ga;                                  // global_addr[31:0]
  g0[3] = (unsigned)((ga >> 32) & 0x1FFFFFFull) | (2u << 30); // global_addr[56:32], type=2
  i32x8 g1;
  g1[0] = (int)(2u << 16);                 // workgroup_mask=0, data_size=2 (4 bytes)
  g1[1] = (int)((unsigned)TILE_ELEMS << 16); // tensor_dim0[15:0] @ bits 63:48
  g1[2] = (int)(1u << 16);                 // tensor_dim1 = 1 @ bits 95:80
  g1[3] = (int)((unsigned)TILE_ELEMS << 16); // tile_dim0 @ bits 127:112
  g1[4] = 1;                               // tile_dim1 = 1, tile_dim2 = 0
  g1[5] = TILE_ELEMS;                      // tensor_dim0_stride[31:0]
  g1[6] = 0;
  g1[7] = 0;
  i32x4 z4 = {};
#if __clang_major__ >= 23
  i32x8 z8 = {};
  __builtin_amdgcn_tensor_load_to_lds(g0, g1, z4, z4, z8, 0);
#else
  __builtin_amdgcn_tensor_load_to_lds(g0, g1, z4, z4, 0);
#endif
#endif
}

__device__ __forceinline__ void tdm_wait_all() {
#ifdef USE_TDM_PATH
  __builtin_amdgcn_s_wait_tensorcnt(0);
#endif
}

#ifndef USE_TDM_PATH
// fallback: cooperative vectorized copy (TILE_ELEMS % 4 == 0)
__device__ __forceinline__ void copy_tile(float* dst, const float* src, int tid) {
  const float4* s = (const float4*)src;
  float4* d = (float4*)dst;
  for (int idx = tid; idx < TILE_ELEMS / 4; idx += 256) d[idx] = s[idx];
}
#endif

// ---------------------------------------------------------------- zero accum
__global__ void k_zero(float* acc) {
  if (threadIdx.x < 8) acc[threadIdx.x] = 0.f;
}

// ------------------------------------------- relu(P_nmf/N_nmf) + column norms
__global__ __launch_bounds__(256) void k_prep(const float* __restrict__ Pn,
                                              const float* __restrict__ Nn,
                                              float* __restrict__ P,
                                              float* __restrict__ N,
                                              float* __restrict__ normP,
                                              float* __restrict__ normN) {
  __shared__ float red[256];
  int b = blockIdx.x;                       // 0..119
  const float* src = (b < STYLE_NUM) ? Pn : Nn;
  float* dst       = (b < STYLE_NUM) ? P  : N;
  float* nrm       = (b < STYLE_NUM) ? normP : normN;
  int col = (b < STYLE_NUM) ? b : b - STYLE_NUM;
  float s = 0.f;
  for (int k = threadIdx.x; k < DIM2; k += blockDim.x) {
    float v = fmaxf(src[k * STYLE_NUM + col], 0.f);
    dst[k * STYLE_NUM + col] = v;
    s += v * v;
  }
  red[threadIdx.x] = s;
  __syncthreads();
  for (int o = 128; o > 0; o >>= 1) {
    if (threadIdx.x < o) red[threadIdx.x] += red[threadIdx.x + o];
    __syncthreads();
  }
  if (threadIdx.x == 0) nrm[col] = red[0];
}

// ------------------------------------------------ per-style embedding scores
__global__ __launch_bounds__(256) void k_style(const float* __restrict__ P,
                                               const float* __restrict__ N,
                                               const float* __restrict__ Wt,
                                               const float* __restrict__ bt,
                                               const float* __restrict__ Wb,
                                               const float* __restrict__ bb,
                                               float* __restrict__ scoreP,
                                               float* __restrict__ scoreN) {
  __shared__ float red[256];
  int b = blockIdx.x;                       // 0..119
  const float* M = (b < STYLE_NUM) ? P : N;
  float* out     = (b < STYLE_NUM) ? scoreP : scoreN;
  int s = (b < STYLE_NUM) ? b : b - STYLE_NUM;
  int h = threadIdx.x;
  float zt = bt[h], zb = bb[h];
  for (int k = 0; k < INPUT_DIM; ++k) {
    zt += M[k * STYLE_NUM + s] * Wt[k * HIDDEN + h];
    zb += M[(k + INPUT_DIM) * STYLE_NUM + s] * Wb[k * HIDDEN + h];
  }
  red[h] = sigm(zt) * sigm(zb);
  __syncthreads();
  for (int o = 128; o > 0; o >>= 1) {
    if (h < o) red[h] += red[h + o];
    __syncthreads();
  }
  if (h == 0) out[s] = red[0];
}

// --------------------- fused Ti/Bj/Bk GEMMs + sigmoid + row dots + loss1/auc
// block = 256 thr (8 waves), 16 batch rows / block; wave w owns hidden cols [32w,32w+32)
// A tiles staged to LDS via TDM; inner loop has zero branches.
__global__ __launch_bounds__(256) void k_emb(const float* __restrict__ iF,
                                             const float* __restrict__ jF,
                                             const float* __restrict__ kF,
                                             const float* __restrict__ Wt,
                                             const float* __restrict__ bt,
                                             const float* __restrict__ Wb,
                                             const float* __restrict__ bb,
                                             float* __restrict__ acc) {
  __shared__ __align__(16) float tI[TILE_ELEMS];
  __shared__ __align__(16) float tJ[TILE_ELEMS];
  __shared__ __align__(16) float tK[TILE_ELEMS];
  __shared__ float s_ij[16], s_ik[16], s_l1[16], s_au[16];
  int tid = threadIdx.x;
  int lane = tid & 31, w = tid >> 5;
  int half = (lane < 16) ? 0 : 2;
  int lr = lane & 15;
  int m0 = blockIdx.x * 16;

  if (tid < 16) { s_ij[tid] = 0.f; s_ik[tid] = 0.f; }
#ifdef USE_TDM_PATH
  if (w == 0) {
    tdm_issue_tile(tI, iF + (long)m0 * INPUT_DIM);
    tdm_issue_tile(tJ, jF + (long)m0 * INPUT_DIM);
    tdm_issue_tile(tK, kF + (long)m0 * INPUT_DIM);
    tdm_wait_all();
  }
#else
  copy_tile(tI, iF + (long)m0 * INPUT_DIM, tid);
  copy_tile(tJ, jF + (long)m0 * INPUT_DIM, tid);
  copy_tile(tK, kF + (long)m0 * INPUT_DIM, tid);
#endif
  __syncthreads();

  int rowOff = lr * INPUT_DIM;
  int c0 = w * 32 + lr, c1 = c0 + 16;

  v8f ti0 = {}, ti1 = {}, aj0 = {}, aj1 = {}, ak0 = {}, ak1 = {};
  // main loop: 90 fully-unguarded K-steps (k0 = 0..356)
  for (int k0 = 0; k0 < 360; k0 += 4) {
    int kk = k0 + half;
    v2f ai = {tI[rowOff + kk], tI[rowOff + kk + 1]};
    v2f aj = {tJ[rowOff + kk], tJ[rowOff + kk + 1]};
    v2f ak = {tK[rowOff + kk], tK[rowOff + kk + 1]};
    v2f wt0 = {Wt[kk * HIDDEN + c0], Wt[(kk + 1) * HIDDEN + c0]};
    v2f wt1 = {Wt[kk * HIDDEN + c1], Wt[(kk + 1) * HIDDEN + c1]};
    v2f wb0 = {Wb[kk * HIDDEN + c0], Wb[(kk + 1) * HIDDEN + c0]};
    v2f wb1 = {Wb[kk * HIDDEN + c1], Wb[(kk + 1) * HIDDEN + c1]};
    ti0 = wmma4(ai, wt0, ti0);
    ti1 = wmma4(ai, wt1, ti1);
    aj0 = wmma4(aj, wb0, aj0);
    aj1 = wmma4(aj, wb1, aj1);
    ak0 = wmma4(ak, wb0, ak0);
    ak1 = wmma4(ak, wb1, ak1);
  }
  // epilogue k0 = 360: lanes>=16 have kk=362/363 (OOB). Load from the clamped
  // (always-valid) address unconditionally, then multiply by a 0/1 mask —
  // branch-free (no EXEC-predicated loads).
  {
    int kk = 360 + half;
    bool v0 = kk < INPUT_DIM;
    int kc = v0 ? kk : 360;                 // always-valid address
    float mk = v0 ? 1.f : 0.f;              // zero-mask as a multiplier
    v2f ai = {tI[rowOff + kc] * mk, tI[rowOff + kc + 1] * mk};
    v2f aj = {tJ[rowOff + kc] * mk, tJ[rowOff + kc + 1] * mk};
    v2f ak = {tK[rowOff + kc] * mk, tK[rowOff + kc + 1] * mk};
    v2f wt0 = {Wt[kc * HIDDEN + c0] * mk, Wt[(kc + 1) * HIDDEN + c0] * mk};
    v2f wt1 = {Wt[kc * HIDDEN + c1] * mk, Wt[(kc + 1) * HIDDEN + c1] * mk};
    v2f wb0 = {Wb[kc * HIDDEN + c0] * mk, Wb[(kc + 1) * HIDDEN + c0] * mk};
    v2f wb1 = {Wb[kc * HIDDEN + c1] * mk, Wb[(kc + 1) * HIDDEN + c1] * mk};
    ti0 = wmma4(ai, wt0, ti0);
    ti1 = wmma4(ai, wt1, ti1);
    aj0 = wmma4(aj, wb0, aj0);
    aj1 = wmma4(aj, wb1, aj1);
    ak0 = wmma4(ak, wb0, ak0);
    ak1 = wmma4(ak, wb1, ak1);
  }

  float bt0 = bt[c0], bt1 = bt[c1], bb0 = bb[c0], bb1 = bb[c1];
#pragma unroll
  for (int v = 0; v < 8; ++v) {
    float t0 = sigm(ti0[v] + bt0), t1 = sigm(ti1[v] + bt1);
    float j0 = sigm(aj0[v] + bb0), j1 = sigm(aj1[v] + bb1);
    float q0 = sigm(ak0[v] + bb0), q1 = sigm(ak1[v] + bb1);
    int row = (lane < 16) ? v : v + 8;
    atomicAdd(&s_ij[row], t0 * j0 + t1 * j1);
    atomicAdd(&s_ik[row], t0 * q0 + t1 * q1);
  }
  __syncthreads();
  if (tid < 16) {
    float x = s_ij[tid] - s_ik[tid];
    s_l1[tid] = softplus(-x);
    s_au[tid] = (x >= 0.f) ? 1.f : 0.f;
  }
  __syncthreads();
  if (tid == 0) {
    float l1 = 0.f, au = 0.f;
    for (int r = 0; r < 16; ++r) { l1 += s_l1[r]; au += s_au[r]; }
    atomicAdd(&acc[0], l1);
    atomicAdd(&acc[3], au);
  }
}

// ------------------- nearest-style argmin (||o||^2 drops) + loss2 accum
// block = 256 thr, 16 rows; waves 0-3 -> P tiles of 16 styles, waves 4-7 -> N
__global__ __launch_bounds__(256) void k_argmin(const float* __restrict__ iF,
                                                const float* __restrict__ jF,
                                                const float* __restrict__ kF,
                                                const float* __restrict__ P,
                                                const float* __restrict__ N,
                                                const float* __restrict__ normP,
                                                const float* __restrict__ normN,
                                                const float* __restrict__ scoreP,
                                                const float* __restrict__ scoreN,
                                                float* __restrict__ acc) {
  __shared__ __align__(16) float tI[TILE_ELEMS];
  __shared__ __align__(16) float tJ[TILE_ELEMS];
  __shared__ __align__(16) float tK[TILE_ELEMS];
  __shared__ float d2[2][16][64];
  __shared__ float sSc[2][16];
  __shared__ float red[16];
  int tid = threadIdx.x, lane = tid & 31, w = tid >> 5;
  int mi = (w < 4) ? 0 : 1;
  int n0 = (w & 3) * 16;
  int half = (lane < 16) ? 0 : 2;
  int lr = lane & 15;
  int m0 = blockIdx.x * 16;

#ifdef USE_TDM_PATH
  if (w == 0) {
    tdm_issue_tile(tI, iF + (long)m0 * INPUT_DIM);
    tdm_issue_tile(tJ, jF + (long)m0 * INPUT_DIM);
    tdm_issue_tile(tK, kF + (long)m0 * INPUT_DIM);
    tdm_wait_all();
  }
#else
  copy_tile(tI, iF + (long)m0 * INPUT_DIM, tid);
  copy_tile(tJ, jF + (long)m0 * INPUT_DIM, tid);
  copy_tile(tK, kF + (long)m0 * INPUT_DIM, tid);
#endif
  __syncthreads();

  const float* tO  = (mi == 0) ? tJ : tK;
  const float* S   = (mi == 0) ? P : N;
  const float* nrm = (mi == 0) ? normP : normN;
  int col = n0 + lr;
  bool cval = col < STYLE_NUM;
  int colc = cval ? col : 0;                // always-valid; garbage cols discarded
  int rowOff = lr * INPUT_DIM;

  v8f d = {};
  for (int k0 = 0; k0 < DIM2; k0 += 4) {    // 724 = 181*4, no tail; pairs never
    int kk = k0 + half;                     // straddle the i|j boundary (362 even)
    const float* ap = (kk < INPUT_DIM) ? (tI + rowOff + kk)
                                       : (tO + rowOff + kk - INPUT_DIM);
    v2f a = {ap[0], ap[1]};
    v2f b = {S[kk * STYLE_NUM + colc], S[(kk + 1) * STYLE_NUM + colc]};
    d = wmma4(a, b, d);
  }
  float nv = nrm[colc];
#pragma unroll
  for (int v = 0; v < 8; ++v) {
    int row = (lane < 16) ? v : v + 8;
    d2[mi][row][col] = cval ? (nv - 2.f * d[v]) : 3.0e38f;
  }
  __syncthreads();
  if (tid < 32) {                           // 2 matrices x 16 rows
    int m = tid >> 4, r = tid & 15;
    float best = d2[m][r][0];
    int bi = 0;
    for (int s = 1; s < STYLE_NUM; ++s) {
      float vv = d2[m][r][s];
      if (vv < best) { best = vv; bi = s; } // first-min like jnp.argmin
    }
    sSc[m][r] = ((m == 0) ? scoreP : scoreN)[bi];
  }
  __syncthreads();
  if (tid < 16) {
    float x = sSc[0][tid] - sSc[1][tid];
    red[tid] = softplus(-x);
  }
  __syncthreads();
  if (tid == 0) {
    float l2 = 0.f;
    for (int r = 0; r < 16; ++r) l2 += red[r];
    atomicAdd(&acc[1], l2);
  }
}

// ------- fused loss3: sum over (724 x 49710) of (pos-P@H1)^2 + (neg-N@H2)^2
// wave owns one 16-col strip; H1/H2 strips cached in regs for full K=60;
// loops all 46 M-tiles so pos/neg stream from HBM exactly once.
__global__ __launch_bounds__(256) void k_nmf(const float* __restrict__ P,
                                             const float* __restrict__ N,
                                             const float* __restrict__ H1,
                                             const float* __restrict__ H2,
                                             const float* __restrict__ pos,
                                             const float* __restrict__ neg,
                                             float* __restrict__ acc) {
  __shared__ float red[256];
  int tid = threadIdx.x, lane = tid & 31, w = tid >> 5;
  int half = (lane < 16) ? 0 : 2;
  int lr = lane & 15;
  long n0 = (long)blockIdx.x * 128 + w * 16;
  long col = n0 + lr;
  bool nval = col < OUTFIT_NUM;
  long cl = nval ? col : 0;                 // always-valid column address

  v2f B1[15], B2[15];
#pragma unroll
  for (int s = 0; s < 15; ++s) {            // K = 60 = 15*4 exactly
    int kk = 4 * s + half;
    B1[s].x = fmaxf(H1[(long)kk * OUTFIT_NUM + cl], 0.f);
    B1[s].y = fmaxf(H1[(long)(kk + 1) * OUTFIT_NUM + cl], 0.f);
    B2[s].x = fmaxf(H2[(long)kk * OUTFIT_NUM + cl], 0.f);
    B2[s].y = fmaxf(H2[(long)(kk + 1) * OUTFIT_NUM + cl], 0.f);
  }

  float local = 0.f;
  for (int mt = 0; mt < 46; ++mt) {         // 46*16 = 736 >= 724 rows
    int arow = mt * 16 + lr;
    int ar = (arow < DIM2) ? arow : (DIM2 - 1);  // clamp; garbage rows discarded
    v8f a1 = {}, a2 = {};
#pragma unroll
    for (int s = 0; s < 15; ++s) {
      int kk = 4 * s + half;
      v2f aP = {P[ar * STYLE_NUM + kk], P[ar * STYLE_NUM + kk + 1]};
      v2f aN = {N[ar * STYLE_NUM + kk], N[ar * STYLE_NUM + kk + 1]};
      a1 = wmma4(aP, B1[s], a1);
      a2 = wmma4(aN, B2[s], a2);
    }
#pragma unroll
    for (int v = 0; v < 8; ++v) {
      int r = mt * 16 + ((lane < 16) ? v : v + 8);
      int rc = (r < DIM2) ? r : (DIM2 - 1);
      float pv = pos[(long)rc * OUTFIT_NUM + cl];
      float nv = neg[(long)rc * OUTFIT_NUM + cl];
      float dp = pv - a1[v];
      float dn = nv - a2[v];
      float contrib = dp * dp + dn * dn;
      local += (r < DIM2 && nval) ? contrib : 0.f;
    }
  }
  red[tid] = local;
  __syncthreads();
  for (int o = 128; o > 0; o >>= 1) {
    if (tid < o) red[tid] += red[tid + o];
    __syncthreads();
  }
  if (tid == 0) atomicAdd(&acc[2], red[0]);
}

// ---------------------------------------------------------------- finalize
__global__ void k_final(const float* __restrict__ acc, float* __restrict__ out) {
  if (threadIdx.x == 0) {
    float l1 = acc[0] / (float)BATCH;
    float l2 = acc[1] / (float)BATCH;
    float l3 = acc[2] / ((float)DIM2 * (float)OUTFIT_NUM);
    float auc = acc[3] / (float)BATCH;
    out[0] = l1 + l2 + 0.1f * l3;
    out[1] = l1;
    out[2] = l2;
    out[3] = l3;
    out[4] = auc;
  }
}

extern "C" void kernel_launch(void* const* d_in, const int* in_sizes, int n_in,
                              void* d_out, int out_size, void* d_ws, size_t ws_size,
                              hipStream_t stream) {
  (void)in_sizes; (void)n_in; (void)out_size; (void)ws_size;
  const float* iF  = (const float*)d_in[0];
  const float* jF  = (const float*)d_in[1];
  const float* kF  = (const float*)d_in[2];
  const float* Wt  = (const float*)d_in[3];
  const float* bt  = (const float*)d_in[4];
  const float* Wb  = (const float*)d_in[5];
  const float* bb  = (const float*)d_in[6];
  const float* Pn  = (const float*)d_in[7];
  const float* Nn  = (const float*)d_in[8];
  const float* H1  = (const float*)d_in[9];
  const float* H2  = (const float*)d_in[10];
  const float* pos = (const float*)d_in[11];
  const float* neg = (const float*)d_in[12];

  float* ws = (float*)d_ws;
  float* P      = ws;                 // 724*60
  float* N      = ws + 43440;         // 724*60
  float* normP  = ws + 86880;         // 64
  float* normN  = ws + 86944;         // 64
  float* scoreP = ws + 87008;         // 64
  float* scoreN = ws + 87072;         // 64
  float* acc    = ws + 87136;         // [loss1, loss2, loss3, auc]
  float* out = (float*)d_out;

  k_zero<<<1, 32, 0, stream>>>(acc);
  k_prep<<<120, 256, 0, stream>>>(Pn, Nn, P, N, normP, normN);
  k_style<<<120, 256, 0, stream>>>(P, N, Wt, bt, Wb, bb, scoreP, scoreN);
  k_emb<<<BATCH / 16, 256, 0, stream>>>(iF, jF, kF, Wt, bt, Wb, bb, acc);
  k_argmin<<<BATCH / 16, 256, 0, stream>>>(iF, jF, kF, P, N, normP, normN,
                                           scoreP, scoreN, acc);
  k_nmf<<<(OUTFIT_NUM + 127) / 128, 256, 0, stream>>>(P, N, H1, H2, pos, neg, acc);
  k_final<<<1, 1, 0, stream>>>(acc, out);
}